// QuantumContrastiveModel_30855045055025
// MI455X (gfx1250) — compile-verified
//
#include <hip/hip_runtime.h>
#include <math.h>

// ---------------------------------------------------------------------------
// QuantumContrastiveModel on MI455X (gfx1250, wave32, WMMA).
//
// Math: StronglyEntanglingLayers is a unitary U applied with the SAME theta to
// both amplitude-embedded states, so <U va|U vb> = va.vb (real unit vectors)
// => fid = clip((va.vb)^2, 0, 1); theta is dead. Fused kernel:
//   v = normalize(tanh(nan_to_num(relu(x@W1+b1)@W2+b2))) for both images,
//   out = (va.vb)^2 clamped.
//
// HBM-bound (411 MB image reads -> ~17.6 us @ 23.3 TB/s), one streaming pass,
// fp32 WMMA 16x16x4 (down-converting saves no bytes). W1/W2 are repacked in
// LDS into WMMA-B "pair" layout so every B-fragment is ONE aligned
// ds_load_b64 straight into an even VGPR pair (no v_mov shuffling):
//   pairRow pr holds (W[2pr][n], W[2pr+1][n]) at dword pr*96 + 2n.
// Row stride 96 dwords => pair-rows pr and pr+1 land in opposite 32-bank
// halves, so lanes 0-15 (pr) and 16-31 (pr+1) never conflict.
// ---------------------------------------------------------------------------

typedef __attribute__((ext_vector_type(2))) float v2f;
typedef __attribute__((ext_vector_type(8))) float v8f;

#define WAVES  8
#define K1     784
#define N1     64
#define N2     16
#define PR1    (K1 / 2)     // 392 W1 pair-rows
#define PR2    (N1 / 2)     // 32  W2 pair-rows
#define PSTR   96           // pair-row stride in dwords (48 v2f)
#define PSTRV  (PSTR / 2)   // pair-row stride in v2f units
#define H_LD   68           // padded h-tile row stride (dwords)

#define LDS_W1 (PR1 * PSTR)                 // 37632 floats
#define LDS_W2 (PR2 * PSTR)                 // 3072 floats
#define LDS_H  (WAVES * 16 * H_LD)          // 8704 floats
// total 49408 floats = 197,632 B < 320 KB/WGP

__device__ __forceinline__ v8f wmma4(v2f a, v2f b, v8f c) {
  // D = A(16x4 f32) * B(4x16 f32) + C(16x16 f32)
  return __builtin_amdgcn_wmma_f32_16x16x4_f32(
      false, a, false, b, (short)0, c, false, false);
}

// Encode one 16-sample tile of one image -> normalized features in C/D layout:
// u[i] = feature (N = lane&15) of sample (base + i + 8*(lane>>4)).
__device__ __forceinline__ v8f encode_tile(
    const float* __restrict__ img, int baseRow,
    const float* __restrict__ sW1p, const float* __restrict__ sW2p,
    float* __restrict__ sH,
    const float* __restrict__ b1, const float* __restrict__ b2, int lane) {
  const int n    = lane & 15;
  const int half = lane >> 4;        // upper half-lanes own K+2,K+3
  const int koff = half * 2;

  // ---- GEMM1: (16 x 784) @ (784 x 64), 4 N-tiles ----
  v8f a0 = {}, a1 = {}, a2 = {}, a3 = {};
  const float* arow = img + (size_t)(baseRow + n) * K1 + koff;
  const v2f* bcol = (const v2f*)sW1p + (half * PSTRV + n);
  for (int k = 0; k < K1; k += 4) {
    v2f av = __builtin_nontemporal_load((const v2f*)(arow + k));  // b64 stream
    const v2f* bp = bcol + (k >> 1) * PSTRV;
    v2f b0 = bp[0];    // (W1[kk][n],    W1[kk+1][n]), kk = k+koff
    v2f b1v = bp[16];  // n+16 tile
    v2f b2v = bp[32];  // n+32 tile
    v2f b3v = bp[48];  // n+48 tile
    a0 = wmma4(av, b0, a0);
    a1 = wmma4(av, b1v, a1);
    a2 = wmma4(av, b2v, a2);
    a3 = wmma4(av, b3v, a3);
  }

  // ---- bias + ReLU, spill h (16x64) to LDS: C/D layout -> A layout ----
  const int mhalf = 8 * half;
  const float bb0 = b1[n], bb1 = b1[16 + n], bb2 = b1[32 + n], bb3 = b1[48 + n];
#pragma unroll
  for (int i = 0; i < 8; ++i) {
    float* hp = sH + (mhalf + i) * H_LD + n;
    hp[0]  = fmaxf(a0[i] + bb0, 0.0f);
    hp[16] = fmaxf(a1[i] + bb1, 0.0f);
    hp[32] = fmaxf(a2[i] + bb2, 0.0f);
    hp[48] = fmaxf(a3[i] + bb3, 0.0f);
  }
  __syncthreads();  // uniform; orders LDS writes before GEMM2 reads

  // ---- GEMM2: (16 x 64) @ (64 x 16) ----
  v8f t = {};
  const float* hrow = sH + n * H_LD + koff;         // b64-aligned, conflict-free
  const v2f* wcol = (const v2f*)sW2p + (half * PSTRV + n);
#pragma unroll
  for (int k = 0; k < N1; k += 4) {
    v2f av = *(const v2f*)(hrow + k);               // ds_load_b64
    v2f bv = wcol[(k >> 1) * PSTRV];                // ds_load_b64
    t = wmma4(av, bv, t);
  }
  __syncthreads();  // sH reusable for the next image

  // ---- epilogue: +b2, nan_to_num, tanh, double normalize ----
  const float bb = b2[n];
  v8f u;
#pragma unroll
  for (int i = 0; i < 8; ++i) {
    float x = t[i] + bb;
    if (__builtin_isnan(x)) x = 0.0f;
    if (__builtin_isinf(x)) x = (x > 0.0f) ? 1.0f : -1.0f;
    u[i] = tanhf(x);
  }
#pragma unroll
  for (int i = 0; i < 8; ++i) {
    float s = u[i] * u[i];          // reduce over the 16 lanes of one sample
    s += __shfl_xor(s, 1, 32);
    s += __shfl_xor(s, 2, 32);
    s += __shfl_xor(s, 4, 32);
    s += __shfl_xor(s, 8, 32);
    u[i] = u[i] / fmaxf(sqrtf(s), 1e-8f);   // F.normalize(eps=1e-8)
    float s2 = u[i] * u[i];                  // AmplitudeEmbedding renormalize
    s2 += __shfl_xor(s2, 1, 32);
    s2 += __shfl_xor(s2, 2, 32);
    s2 += __shfl_xor(s2, 4, 32);
    s2 += __shfl_xor(s2, 8, 32);
    u[i] *= rsqrtf(s2);
  }
  return u;
}

extern "C" __global__ void __launch_bounds__(256, 1)
qcm_fused_kernel(const float* __restrict__ img_a, const float* __restrict__ img_b,
                 const float* __restrict__ W1, const float* __restrict__ b1,
                 const float* __restrict__ W2, const float* __restrict__ b2,
                 const float* __restrict__ theta, float* __restrict__ out) {
  extern __shared__ float smem[];
  float* sW1p = smem;                       // WMMA-B pair layout of W1
  float* sW2p = smem + LDS_W1;              // WMMA-B pair layout of W2
  float* sHall = smem + LDS_W1 + LDS_W2;    // per-wave h tiles

  const int tid  = threadIdx.x;
  const int lane = tid & 31;
  const int wave = tid >> 5;

  // Repack W1 into pair layout: one b128 LDS store per (pair-row, col-pair).
  // dword addr pr*96 + 2c is 16B-aligned for even c (384*pr + 8c).
  for (int t = tid; t < PR1 * 32; t += 256) {
    const int pr = t >> 5;
    const int c  = (t & 31) * 2;
    const float* g = W1 + (size_t)(2 * pr) * N1 + c;
    const float2 r0 = *(const float2*)g;          // W1[2pr][c..c+1]
    const float2 r1 = *(const float2*)(g + N1);   // W1[2pr+1][c..c+1]
    *(float4*)(sW1p + pr * PSTR + 2 * c) = make_float4(r0.x, r1.x, r0.y, r1.y);
  }
  // Repack W2 the same way (32 pair-rows x 8 col-pairs).
  for (int t = tid; t < PR2 * 8; t += 256) {
    const int pr = t >> 3;
    const int c  = (t & 7) * 2;
    const float* g = W2 + 2 * pr * N2 + c;
    const float2 r0 = *(const float2*)g;
    const float2 r1 = *(const float2*)(g + N2);
    *(float4*)(sW2p + pr * PSTR + 2 * c) = make_float4(r0.x, r1.x, r0.y, r1.y);
  }
  __syncthreads();

  float* sH = sHall + wave * (16 * H_LD);
  const int baseRow = (blockIdx.x * WAVES + wave) * 16;

  v8f ua = encode_tile(img_a, baseRow, sW1p, sW2p, sH, b1, b2, lane);
  v8f ub = encode_tile(img_b, baseRow, sW1p, sW2p, sH, b1, b2, lane);

  // fid = clip((ua.ub)^2, 0, 1); unitary circuit cancels, theta dead.
#pragma unroll
  for (int i = 0; i < 8; ++i) {
    float p = ua[i] * ub[i];
    p += __shfl_xor(p, 1, 32);
    p += __shfl_xor(p, 2, 32);
    p += __shfl_xor(p, 4, 32);
    p += __shfl_xor(p, 8, 32);
    if ((lane & 15) == 0) {
      float f = fminf(fmaxf(p * p, 0.0f), 1.0f);
      out[baseRow + i + 8 * (lane >> 4)] = f;
    }
  }
  (void)theta;
}

extern "C" void kernel_launch(void* const* d_in, const int* in_sizes, int n_in,
                              void* d_out, int out_size, void* d_ws, size_t ws_size,
                              hipStream_t stream) {
  const float* img_a = (const float*)d_in[0];
  const float* img_b = (const float*)d_in[1];
  const float* W1    = (const float*)d_in[2];
  const float* b1    = (const float*)d_in[3];
  const float* W2    = (const float*)d_in[4];
  const float* b2    = (const float*)d_in[5];
  const float* theta = (const float*)d_in[6];
  float* out = (float*)d_out;

  const int B = in_sizes[0] / (28 * 28);   // 65536
  const int blocks = B / (WAVES * 16);     // 16 samples per wave, 8 waves

  const size_t shmem = (size_t)(LDS_W1 + LDS_W2 + LDS_H) * sizeof(float);
  // 197,632 bytes — within the 320 KB/workgroup LDS budget of a CDNA5 WGP.

  qcm_fused_kernel<<<dim3(blocks), dim3(256), shmem, stream>>>(
      img_a, img_b, W1, b1, W2, b2, theta, out);
  (void)d_ws; (void)ws_size; (void)n_in; (void)out_size;
}